// Repro_54176717471998
// MI455X (gfx1250) — compile-verified
//
#include <hip/hip_runtime.h>
#include <stdint.h>

// Problem constants (match reference)
#define NUM_ENT 14541
#define DIM     400
#define BQ      32        // B queries
#define N_CAND  14505

#define CAND_TILE 32      // candidates per block
#define WAVES     8       // 256 threads / wave32
#define CPW       4       // candidates per wave (8 waves * 4 = 32 = CAND_TILE)

// LDS layout (dynamic shared only -> group segment offset 0):
//   [0, 102400)      : qo table, d-major float2 {q,o}: qo_t[d*32 + b]
//   [+0, +51200)     : candidate tile, row-major, stride DIM (TDM writes it)
//   [+0, +4224)      : output transpose tile, stride 33 (conflict-free r/w)
#define QO_LDS_BYTES  (BQ * DIM * 8)            // 102400
#define C_LDS_OFF     QO_LDS_BYTES
#define C_LDS_BYTES   (CAND_TILE * DIM * 4)     // 51200
#define OUT_LDS_OFF   (C_LDS_OFF + C_LDS_BYTES)
#define OUT_STRIDE    33
#define OUT_LDS_BYTES (CAND_TILE * OUT_STRIDE * 4)
#define SMEM_BYTES    (OUT_LDS_OFF + OUT_LDS_BYTES) // 157824 (< 320KB WGP LDS)

typedef unsigned int v4u __attribute__((ext_vector_type(4)));
typedef int          v8i __attribute__((ext_vector_type(8)));
typedef int          v4i __attribute__((ext_vector_type(4)));

// ---------------- TDM descriptor helpers (ISA cdna5 §8.3-8.7) ---------------
// Group0: [1:0]=count(1), [30]=gather_index_size(0=16b), [31]=gather_mode,
//         [63:32]=lds_addr(bytes), [120:64]=global_addr, [127:126]=type(2)
__device__ __forceinline__ v4u tdm_group0(uint32_t lds_addr, const void* g,
                                          bool gather) {
  uint64_t ga = (uint64_t)(uintptr_t)g;
  v4u g0;
  g0.x = 1u | (gather ? 0x80000000u : 0u);      // count=1, 16-bit indices
  g0.y = lds_addr;
  g0.z = (uint32_t)ga;
  g0.w = (uint32_t)(ga >> 32) | (2u << 30);     // type=2 ("image")
  return g0;
}

// Group1 bit layout (see §8.4): data_size[17:16], tensor_dim0[79:48],
// tensor_dim1[111:80], tile_dim0[127:112], tile_dim1[143:128],
// tensor_dim0_stride[207:160].
__device__ __forceinline__ v8i tdm_group1(uint32_t tensor_d0, uint32_t tensor_d1,
                                          uint32_t tile_d0, uint32_t tile_d1,
                                          uint32_t d0_stride) {
  v8i g1;
  g1[0] = (int)(2u << 16);                            // data_size=2 -> 4B elems
  g1[1] = (int)((tensor_d0 & 0xFFFFu) << 16);         // tensor_dim0[15:0]
  g1[2] = (int)((tensor_d0 >> 16) | ((tensor_d1 & 0xFFFFu) << 16));
  g1[3] = (int)((tensor_d1 >> 16) | ((tile_d0 & 0xFFFFu) << 16));
  g1[4] = (int)(tile_d1 & 0xFFFFu);                   // tile_dim1 (gather: #idx)
  g1[5] = (int)d0_stride;                             // stride[31:0]
  g1[6] = 0;                                          // stride[47:32], dim1_stride
  g1[7] = 0;
  return g1;
}

__device__ __forceinline__ void tdm_load(v4u g0, v8i g1, v4i g2, v4i g3) {
  // This toolchain exposes the 6-arg form: the 5th (int32x8) group is passed
  // zero-filled (semantics uncharacterized; groups 0-3 carry the full D#).
  v8i gx = {0, 0, 0, 0, 0, 0, 0, 0};
  __builtin_amdgcn_tensor_load_to_lds(g0, g1, g2, g3, gx, 0);
}

// ---- Kernel 1: build d-major interleaved (q = head+rel_c, o = rel_o) table -
__global__ void build_queries(const float* __restrict__ ent,
                              const float* __restrict__ relc,
                              const float* __restrict__ relo,
                              const int*   __restrict__ pairs, // (B,2) row-major
                              float2*      __restrict__ qo)    // qo[d*32 + b]
{
  const int b = blockIdx.x;                 // 0..BQ-1
  const int h = pairs[b * 2 + 0];
  const int r = pairs[b * 2 + 1];
  const float* hrow = ent  + (size_t)h * DIM;
  const float* crow = relc + (size_t)r * DIM;
  const float* orow = relo + (size_t)r * DIM;
  for (int d = threadIdx.x; d < DIM; d += blockDim.x) {
    qo[(size_t)d * BQ + b] = make_float2(hrow[d] + crow[d], orow[d]);
  }
}

// ---- Kernel 2: main scoring kernel -----------------------------------------
__global__ void __launch_bounds__(256, 2)
boxe_score(const float*  __restrict__ ent,      // (NUM_ENT, DIM)
           const int*    __restrict__ cand_idx, // arg5 row 0
           const float2* __restrict__ qo,       // ws table, d-major (DIM*BQ)
           const float*  __restrict__ gamma_p,  // arg3 (1,)
           float*        __restrict__ out)      // (BQ, N_CAND)
{
  extern __shared__ char smem[];
  const float2* s_qo  = (const float2*)smem;                 // qo_t[d*32 + b]
  const float*  s_c   = (const float*)(smem + C_LDS_OFF);    // c[r*DIM + d]
  float*        s_out = (float*)(smem + OUT_LDS_OFF);        // [cand*33 + b]

  const int tid  = (int)threadIdx.x;
  const int lane = tid & 31;                 // lane = query index b
  const int wave = tid >> 5;
  const int tile = (int)blockIdx.x * CAND_TILE;

  // ---- Wave 0 drives the Tensor Data Mover; TDM ignores EXEC and signals
  // only the issuing wave, so exactly one wave issues + waits, then barrier.
  if (tid < 32) {
    // (a) 1-D copy of the whole 102,400B query table: one TDM op.
    //     tensor/tile: 25600 4B elements in a single row.
    tdm_load(tdm_group0(0u, qo, /*gather=*/false),
             tdm_group1(/*t_d0=*/BQ * DIM * 2, /*t_d1=*/1,
                        /*tile_d0=*/BQ * DIM * 2, /*tile_d1=*/0,
                        /*d0_stride=*/BQ * DIM * 2),
             v4i{0, 0, 0, 0}, v4i{0, 0, 0, 0});

    // (b) candidate gather: 2 descriptors x 16 rows, 16-bit row indices.
    //     tensor = entity matrix (NUM_ENT x DIM), row r -> LDS row-major.
#pragma unroll
    for (int half = 0; half < 2; ++half) {
      uint32_t w[8];
#pragma unroll
      for (int k = 0; k < 8; ++k) {
        int c0 = tile + half * 16 + 2 * k;
        int c1 = c0 + 1;
        if (c0 > N_CAND - 1) c0 = N_CAND - 1;   // clamp: extra rows harmless
        if (c1 > N_CAND - 1) c1 = N_CAND - 1;
        w[k] = ((uint32_t)cand_idx[c0] & 0xFFFFu) |
               (((uint32_t)cand_idx[c1] & 0xFFFFu) << 16);
      }
      tdm_load(tdm_group0(C_LDS_OFF + (uint32_t)half * 16 * DIM * 4, ent,
                          /*gather=*/true),
               tdm_group1(/*t_d0=*/DIM, /*t_d1=*/NUM_ENT,
                          /*tile_d0=*/DIM, /*tile_d1=*/16,
                          /*d0_stride=*/DIM),
               v4i{(int)w[0], (int)w[1], (int)w[2], (int)w[3]},
               v4i{(int)w[4], (int)w[5], (int)w[6], (int)w[7]});
    }
    __builtin_amdgcn_s_wait_tensorcnt(0);   // TENSORcnt==0: LDS tiles landed
  }
  __syncthreads();

  const float gamma = gamma_p[0];
  const int r0 = wave * CPW;                 // this wave's 4 candidate rows

  float sd[CPW], sm[CPW];
#pragma unroll
  for (int j = 0; j < CPW; ++j) { sd[j] = 0.0f; sm[j] = 0.0f; }

#pragma unroll 2
  for (int d = 0; d < DIM; d += 2) {
    // per-lane query/offset pairs: lane-consecutive b64, spans all 64 banks
    const float2 a0 = s_qo[(size_t)d * BQ + lane];        // {q_d,   o_d}
    const float2 a1 = s_qo[(size_t)(d + 1) * BQ + lane];  // {q_d+1, o_d+1}
#pragma unroll
    for (int j = 0; j < CPW; ++j) {
      // candidate dims: broadcast b64 (conflict-free by definition)
      const float2 c2 = *(const float2*)(s_c + (size_t)(r0 + j) * DIM + d);
      float t0 = c2.x - a0.x;                // 4 VALU per (cand, dim):
      sd[j] += fabsf(t0);                    //   sub, add(|t|),
      sm[j] += fminf(fabsf(t0), a0.y);       //   min(|t|,o), add
      float t1 = c2.y - a1.x;
      sd[j] += fabsf(t1);
      sm[j] += fminf(fabsf(t1), a1.y);
    }
  }

  // out = gamma - (Sd - Sm) - 0.02*Sm = gamma - Sd + 0.98*Sm   (o >= 0)
  // Stage [cand][query] in padded LDS so global stores are coalesced.
#pragma unroll
  for (int j = 0; j < CPW; ++j) {
    s_out[(size_t)(r0 + j) * OUT_STRIDE + lane] = gamma - sd[j] + 0.98f * sm[j];
  }
  __syncthreads();

  // wave writes 4 query-rows; lane = candidate -> 128B coalesced stores
  const int c = tile + lane;
  if (c < N_CAND) {
#pragma unroll
    for (int j = 0; j < CPW; ++j) {
      const int b = r0 + j;
      out[(size_t)b * N_CAND + c] = s_out[(size_t)lane * OUT_STRIDE + b];
    }
  }
}

// ---- Host launch -----------------------------------------------------------
extern "C" void kernel_launch(void* const* d_in, const int* in_sizes, int n_in,
                              void* d_out, int out_size, void* d_ws, size_t ws_size,
                              hipStream_t stream) {
  const float* ent   = (const float*)d_in[0];  // arg0: (NUM_ENT, D)
  const float* relc  = (const float*)d_in[1];  // arg1: (NUM_REL, D)
  const float* relo  = (const float*)d_in[2];  // arg2: (NUM_REL, D)
  const float* gamma = (const float*)d_in[3];  // arg3: (1,)
  const int*   pairs = (const int*)d_in[4];    // arg4: (B, 2)
  const int*   cidx  = (const int*)d_in[5];    // arg5: (4, N_CAND); row 0 used
  float*       out   = (float*)d_out;          // (B, N_CAND)
  float2*      qo    = (float2*)d_ws;          // DIM*BQ float2 = 102400 B

  build_queries<<<BQ, 128, 0, stream>>>(ent, relc, relo, pairs, qo);

  const int grid = (N_CAND + CAND_TILE - 1) / CAND_TILE;  // 454
  (void)hipFuncSetAttribute((const void*)boxe_score,
                            hipFuncAttributeMaxDynamicSharedMemorySize,
                            (int)SMEM_BYTES);
  boxe_score<<<grid, 256, SMEM_BYTES, stream>>>(ent, cidx, qo, gamma, out);
}